// ParallelEuclideanEmbed_35115652612660
// MI455X (gfx1250) — compile-verified
//
#include <hip/hip_runtime.h>

#define B_    16
#define T_    512
#define IN_   8
#define H_    64
#define E_    32
#define NL_   3
#define DI_   128
#define N_    16
#define K_    4
#define DTR_  4
#define XD_   (DTR_ + 2 * N_)   // 36 (logical)
#define XDP_  48                // padded to a multiple of 16 for guard-free WMMA
#define MROWS (B_ * T_)         // 8192

typedef __attribute__((ext_vector_type(2))) float v2f;
typedef __attribute__((ext_vector_type(8))) float v8f;

__device__ __forceinline__ float silu_f(float x) { return x / (1.0f + __expf(-x)); }
__device__ __forceinline__ float softplus_f(float x) {
    return (x > 20.0f) ? x : log1pf(__expf(x));
}

// ---------------------------------------------------------------------------
// fp32 GEMM on V_WMMA_F32_16X16X4_F32: C[M,NV] = A[M,KV] @ W[KV,NV] (+bias).
// W is pre-packed column-major (Wcm[n*KV + k]) so both the A and B operands of
// each WMMA come from a single contiguous global_load_b64 per lane.
// KV, NV compile-time -> fully unrolled K-loop, immediate-offset store clause.
// One wave per 16x16 tile, tn-major so the 8 waves of a block share an M-tile.
// ---------------------------------------------------------------------------
template <int KV, int NV>
__global__ void gemm_wmma_f32(const float* __restrict__ A,
                              const float* __restrict__ Wcm,
                              const float* __restrict__ bias,
                              float* __restrict__ C,
                              int M) {
    const int lane = threadIdx.x & 31;
    const int wave = threadIdx.x >> 5;
    const int tilesM = M >> 4;
    constexpr int tilesN = NV / 16;
    const int tile = blockIdx.x * 8 + wave;
    if (tile >= tilesM * tilesN) return;       // wave-uniform exit (EXEC stays full)

    const int tn = tile % tilesN;              // waves in a block: consecutive tn,
    const int tm = tile / tilesN;              // same tm -> shared A lines in WGP$
    const int half = lane >> 4;                // 0: lanes 0-15, 1: lanes 16-31
    const int l16  = lane & 15;
    const int m0 = tm * 16;
    const int mrow = m0 + l16;                 // A-matrix row for this lane
    const int ncol = tn * 16 + l16;            // B/C/D column for this lane

    const float* __restrict__ Ap = A   + (size_t)mrow * KV + half * 2;
    const float* __restrict__ Wp = Wcm + (size_t)ncol * KV + half * 2;

    v8f acc = {0.f, 0.f, 0.f, 0.f, 0.f, 0.f, 0.f, 0.f};

#pragma unroll
    for (int k = 0; k < KV; k += 4) {
        const v2f a = *reinterpret_cast<const v2f*>(Ap + k);   // A[m][kk,kk+1]
        const v2f b = *reinterpret_cast<const v2f*>(Wp + k);   // W[kk,kk+1][n]
        acc = __builtin_amdgcn_wmma_f32_16x16x4_f32(
            /*neg_a=*/false, a, /*neg_b=*/false, b,
            /*c_mod=*/(short)0, acc, /*reuse_a=*/false, /*reuse_b=*/false);
    }

    const float bv = bias ? bias[ncol] : 0.0f;
    float* __restrict__ Cp = C + (size_t)(m0 + half * 8) * NV + ncol;
#pragma unroll
    for (int r = 0; r < 8; ++r) {              // D VGPR r: M=r (lo half) / r+8 (hi)
        Cp[r * NV] = acc[r] + bv;              // compile-time immediate offsets
    }
}

// ---------------------------------------------------------------------------
// Pack W[K x Nsrc] (row-major) into Wcm[Ndst x K] (column-major), zero-padding
// columns Nsrc..Ndst-1. Weights are tiny (<=64KB) - this is launch noise.
// ---------------------------------------------------------------------------
__global__ void pack_w_k(const float* __restrict__ w, float* __restrict__ wp,
                         int K, int Nsrc, int Ndst) {
    const int idx = blockIdx.x * blockDim.x + threadIdx.x;
    if (idx >= K * Ndst) return;
    const int n = idx / K;
    const int k = idx % K;
    wp[idx] = (n < Nsrc) ? w[(size_t)k * Nsrc + n] : 0.0f;
}

// ---------------------------------------------------------------------------
// Depthwise causal conv (K=4) over the x-half of xz, + bias + SiLU.
// ---------------------------------------------------------------------------
__global__ void conv_silu_k(const float* __restrict__ xz,
                            const float* __restrict__ cw,
                            const float* __restrict__ cb,
                            float* __restrict__ xc) {
    const int idx = blockIdx.x * blockDim.x + threadIdx.x;
    if (idx >= MROWS * DI_) return;
    const int d = idx % DI_;
    const int row = idx / DI_;
    const int t = row % T_;
    float acc = cb[d];
#pragma unroll
    for (int k = 0; k < K_; ++k) {
        const int dtt = k - (K_ - 1);          // -3..0
        if (t + dtt >= 0)
            acc += xz[(size_t)(row + dtt) * (2 * DI_) + d] * cw[d * K_ + k];
    }
    xc[idx] = silu_f(acc);
}

// ---------------------------------------------------------------------------
// dt = softplus(x_dbl[:, :4] @ dt_w + dt_b)      (x_dbl has padded stride 48)
// ---------------------------------------------------------------------------
__global__ void dt_k(const float* __restrict__ xdbl,
                     const float* __restrict__ dtw,
                     const float* __restrict__ dtb,
                     float* __restrict__ dt) {
    const int idx = blockIdx.x * blockDim.x + threadIdx.x;
    if (idx >= MROWS * DI_) return;
    const int d = idx % DI_;
    const int row = idx / DI_;
    float acc = dtb[d];
#pragma unroll
    for (int j = 0; j < DTR_; ++j)
        acc += xdbl[(size_t)row * XDP_ + j] * dtw[j * DI_ + d];
    dt[idx] = softplus_f(acc);
}

// ---------------------------------------------------------------------------
// Selective scan. One block per batch element; 256 threads = 128 channels x 2,
// each thread owns 8 of the N=16 states. B_t/C_t staged into LDS via the CDNA5
// async global->LDS path (ASYNCcnt), combined with a wave32 shfl_xor.
// ---------------------------------------------------------------------------
__global__ void scan_k(const float* __restrict__ dt,
                       const float* __restrict__ xc,
                       const float* __restrict__ xdbl,
                       const float* __restrict__ A_log,
                       float* __restrict__ y) {
    __shared__ float sBC[2 * N_];
    const int b = blockIdx.x;
    const int tid = threadIdx.x;
    const int d = tid >> 1;
    const int n0 = (tid & 1) * 8;

    float Aval[8], hst[8];
#pragma unroll
    for (int j = 0; j < 8; ++j) {
        Aval[j] = -__expf(A_log[d * N_ + n0 + j]);
        hst[j] = 0.0f;
    }

    for (int t = 0; t < T_; ++t) {
        const int row = b * T_ + t;
        __syncthreads();
        if (tid < 2 * N_) {
            // Async DMA 32 floats (B_t|C_t) into LDS; flat LDS address truncates
            // to the LDS byte offset per ISA 10.2 aperture rules.
            unsigned ldsOff = (unsigned)(size_t)(&sBC[tid]);
            const float* gp = &xdbl[(size_t)row * XDP_ + DTR_ + tid];
            asm volatile("global_load_async_to_lds_b32 %0, %1, off\n\t"
                         "s_wait_asynccnt 0x0"
                         :: "v"(ldsOff), "v"(gp)
                         : "memory");
        }
        __syncthreads();

        const float dtv = dt[(size_t)row * DI_ + d];
        const float xv  = xc[(size_t)row * DI_ + d];
        const float dtx = dtv * xv;
        float part = 0.0f;
#pragma unroll
        for (int j = 0; j < 8; ++j) {
            const float dA = __expf(dtv * Aval[j]);
            hst[j] = dA * hst[j] + dtx * sBC[n0 + j];
            part += hst[j] * sBC[N_ + n0 + j];
        }
        const float tot = part + __shfl_xor(part, 1, 32);
        if ((tid & 1) == 0) y[(size_t)row * DI_ + d] = tot;

        if (t + 1 < T_) {                       // hide next step's latency
            __builtin_prefetch(&dt[(size_t)(row + 1) * DI_ + d], 0, 1);
            __builtin_prefetch(&xc[(size_t)(row + 1) * DI_ + d], 0, 1);
        }
    }
}

// ---------------------------------------------------------------------------
// y = (y + xc * Dskip) * silu(z)   (in place on y)
// ---------------------------------------------------------------------------
__global__ void gate_k(float* __restrict__ y,
                       const float* __restrict__ xc,
                       const float* __restrict__ xz,
                       const float* __restrict__ Dsk) {
    const int idx = blockIdx.x * blockDim.x + threadIdx.x;
    if (idx >= MROWS * DI_) return;
    const int d = idx % DI_;
    const int row = idx / DI_;
    const float z = xz[(size_t)row * (2 * DI_) + DI_ + d];
    y[idx] = (y[idx] + xc[idx] * Dsk[d]) * silu_f(z);
}

__global__ void mean_k(const float* __restrict__ h, float* __restrict__ hm) {
    const int idx = blockIdx.x * blockDim.x + threadIdx.x;
    if (idx >= B_ * H_) return;
    const int hc = idx % H_;
    const int b = idx / H_;
    float s = 0.0f;
    for (int t = 0; t < T_; ++t) s += h[(size_t)(b * T_ + t) * H_ + hc];
    hm[idx] = s * (1.0f / (float)T_);
}

__global__ void head_k(const float* __restrict__ hm,
                       const float* __restrict__ Wo,
                       const float* __restrict__ bo,
                       float* __restrict__ out) {
    const int idx = blockIdx.x * blockDim.x + threadIdx.x;
    if (idx >= B_ * E_) return;
    const int e = idx % E_;
    const int b = idx / E_;
    float s = bo[e];
    for (int hc = 0; hc < H_; ++hc) s += hm[b * H_ + hc] * Wo[hc * E_ + e];
    out[idx] = s;
}

__global__ void sum_k(float* __restrict__ out) {
    const int i = blockIdx.x * blockDim.x + threadIdx.x;
    if (i >= B_ * E_) return;
    float s = 0.0f;
    for (int br = 0; br < 4; ++br) s += out[br * B_ * E_ + i];
    out[4 * B_ * E_ + i] = s;
}

// ---------------------------------------------------------------------------
extern "C" void kernel_launch(void* const* d_in, const int* in_sizes, int n_in,
                              void* d_out, int out_size, void* d_ws, size_t ws_size,
                              hipStream_t stream) {
    const float* xin[4] = {(const float*)d_in[0], (const float*)d_in[1],
                           (const float*)d_in[2], (const float*)d_in[3]};
    const float* W_in   = (const float*)d_in[4];
    const float* b_in   = (const float*)d_in[5];
    const float* in_w   = (const float*)d_in[6];
    const float* conv_w = (const float*)d_in[7];
    const float* conv_b = (const float*)d_in[8];
    const float* xproj  = (const float*)d_in[9];
    const float* dt_w   = (const float*)d_in[10];
    const float* dt_b   = (const float*)d_in[11];
    const float* A_log  = (const float*)d_in[12];
    const float* Dskip  = (const float*)d_in[13];
    const float* out_w  = (const float*)d_in[14];
    const float* W_out  = (const float*)d_in[15];
    const float* b_out  = (const float*)d_in[16];
    float* out = (float*)d_out;

    float* ws   = (float*)d_ws;
    float* h    = ws; ws += MROWS * H_;        // 0.5M floats
    float* xz   = ws; ws += MROWS * 2 * DI_;   // 2M
    float* xc   = ws; ws += MROWS * DI_;       // 1M
    float* xdbl = ws; ws += MROWS * XDP_;      // 0.38M (padded stride 48)
    float* dtb  = ws; ws += MROWS * DI_;       // 1M
    float* yb   = ws; ws += MROWS * DI_;       // 1M
    float* wpk  = ws; ws += H_ * 2 * DI_;      // 16K (max packed weight size)
    float* hm   = ws; ws += B_ * H_;

    const int EW = MROWS * DI_;
    const int ewBlocks = (EW + 255) / 256;

    auto pack = [&](const float* w, int K, int Nsrc, int Ndst) {
        pack_w_k<<<(K * Ndst + 255) / 256, 256, 0, stream>>>(w, wpk, K, Nsrc, Ndst);
    };
    auto blocksFor = [&](int M, int NV) { return ((M / 16) * (NV / 16) + 7) / 8; };

    for (int br = 0; br < 4; ++br) {
        // h = x @ W_in + b_in        (8192 x 64 x 8)
        pack(W_in + br * IN_ * H_, IN_, H_, H_);
        gemm_wmma_f32<IN_, H_><<<blocksFor(MROWS, H_), 256, 0, stream>>>(
            xin[br], wpk, b_in + br * H_, h, MROWS);
        for (int l = 0; l < NL_; ++l) {
            const int bl = br * NL_ + l;
            // xz = h @ in_w          (8192 x 256 x 64)
            pack(in_w + (size_t)bl * H_ * 2 * DI_, H_, 2 * DI_, 2 * DI_);
            gemm_wmma_f32<H_, 2 * DI_><<<blocksFor(MROWS, 2 * DI_), 256, 0, stream>>>(
                h, wpk, nullptr, xz, MROWS);
            conv_silu_k<<<ewBlocks, 256, 0, stream>>>(
                xz, conv_w + (size_t)bl * DI_ * K_, conv_b + (size_t)bl * DI_, xc);
            // x_dbl = xc @ xproj    (8192 x 48(pad 36) x 128)
            pack(xproj + (size_t)bl * DI_ * XD_, DI_, XD_, XDP_);
            gemm_wmma_f32<DI_, XDP_><<<blocksFor(MROWS, XDP_), 256, 0, stream>>>(
                xc, wpk, nullptr, xdbl, MROWS);
            dt_k<<<ewBlocks, 256, 0, stream>>>(
                xdbl, dt_w + (size_t)bl * DTR_ * DI_, dt_b + (size_t)bl * DI_, dtb);
            scan_k<<<B_, 256, 0, stream>>>(
                dtb, xc, xdbl, A_log + (size_t)bl * DI_ * N_, yb);
            gate_k<<<ewBlocks, 256, 0, stream>>>(yb, xc, xz, Dskip + (size_t)bl * DI_);
            // h = yg @ out_w         (8192 x 64 x 128)
            pack(out_w + (size_t)bl * DI_ * H_, DI_, H_, H_);
            gemm_wmma_f32<DI_, H_><<<blocksFor(MROWS, H_), 256, 0, stream>>>(
                yb, wpk, nullptr, h, MROWS);
        }
        mean_k<<<(B_ * H_ + 255) / 256, 256, 0, stream>>>(h, hm);
        head_k<<<(B_ * E_ + 255) / 256, 256, 0, stream>>>(
            hm, W_out + (size_t)br * H_ * E_, b_out + (size_t)br * E_,
            out + (size_t)br * B_ * E_);
    }
    sum_k<<<(B_ * E_ + 255) / 256, 256, 0, stream>>>(out);
}